// dap_61357902791335
// MI455X (gfx1250) — compile-verified
//
#include <hip/hip_runtime.h>

#define S3 32
#define NVOX 32768
#define CIN 64
#define CH 32
#define BATCH 2
#define EPSV 1e-5f
#define SLOPE 0.2f

typedef __attribute__((ext_vector_type(2))) float v2f;
typedef __attribute__((ext_vector_type(8))) float v8f;

#if defined(__has_builtin)
#if __has_builtin(__builtin_amdgcn_global_load_async_to_lds_b32)
#define USE_ASYNC_LDS 1
#endif
#endif

typedef __attribute__((address_space(1))) int* gas_i32;
typedef __attribute__((address_space(3))) int* las_i32;

__device__ __forceinline__ float lrelu(float x){ return x > 0.f ? x : SLOPE * x; }

__device__ __forceinline__ void async_wait(){
#ifdef USE_ASYNC_LDS
#if __has_builtin(__builtin_amdgcn_s_wait_asynccnt)
  __builtin_amdgcn_s_wait_asynccnt(0);
#else
  asm volatile("s_wait_asynccnt 0" ::: "memory");
#endif
#endif
}

// ---------------- zero scratch (async OOB target) ----------------
__global__ void k_zero(float* __restrict__ z){ z[threadIdx.x] = 0.f; }

// ---------------- per-channel mean/var over 32^3 voxels ----------------
__global__ void k_stats(const float* __restrict__ in, float* __restrict__ stats){
  int ch = blockIdx.x;
  const float* p = in + (size_t)ch * NVOX;
  float s = 0.f, ss = 0.f;
  for(int i = threadIdx.x; i < NVOX; i += blockDim.x){ float v = p[i]; s += v; ss += v*v; }
  __shared__ float sh[256], sh2[256];
  sh[threadIdx.x] = s; sh2[threadIdx.x] = ss; __syncthreads();
  for(int o = 128; o > 0; o >>= 1){
    if(threadIdx.x < o){ sh[threadIdx.x] += sh[threadIdx.x+o]; sh2[threadIdx.x] += sh2[threadIdx.x+o]; }
    __syncthreads();
  }
  if(threadIdx.x == 0){
    float m = sh[0] * (1.f/NVOX);
    float var = sh2[0] * (1.f/NVOX) - m*m;
    stats[ch*2] = m; stats[ch*2+1] = var;
  }
}

// ---------------- SE gate: fc1 -> lrelu -> fc2 -> softmax ----------------
__global__ void k_gate(const float* __restrict__ statsX,
                       const float* __restrict__ wfc1, const float* __restrict__ bfc1,
                       const float* __restrict__ wfc2, const float* __restrict__ bfc2,
                       float* __restrict__ dw){
  int b = threadIdx.x;
  if(b >= BATCH) return;
  float h[16];
  for(int j = 0; j < 16; j++){
    float a = bfc1[j];
    for(int c = 0; c < CIN; c++) a += wfc1[j*CIN+c] * statsX[(b*CIN+c)*2];
    h[j] = lrelu(a);
  }
  float l0 = bfc2[0], l1 = bfc2[1];
  for(int j = 0; j < 16; j++){ l0 += wfc2[j]*h[j]; l1 += wfc2[16+j]*h[j]; }
  float mx = fmaxf(l0, l1);
  float e0 = __expf(l0-mx), e1 = __expf(l1-mx);
  float inv = 1.f/(e0+e1);
  dw[b*2] = e0*inv; dw[b*2+1] = e1*inv;
}

// -------- pack conv weights into the f32 WMMA A-fragment lane layout --------
// A 16x4 f32 layout: lane m=l%16, h=l/16; VGPR e -> K = e + 2h (per K-step of 4)
__global__ void k_pack(const float* __restrict__ w, float* __restrict__ apack,
                       int mtiles, int mreal){
  int idx = blockIdx.x * blockDim.x + threadIdx.x;
  int total = 27 * mtiles * 8 * 32;
  if(idx >= total) return;
  int lane = idx & 31;
  int s    = (idx >> 5) & 7;
  int mt   = (idx >> 8) % mtiles;
  int t    = idx / (mtiles * 256);
  int m    = mt*16 + (lane & 15);
  int h    = lane >> 4;
  for(int e = 0; e < 2; e++){
    int k = 4*s + 2*h + e;          // input channel within tap
    float v = 0.f;
    if(m < mreal) v = w[((size_t)m * CH + k) * 27 + t];
    apack[(size_t)idx*2 + e] = v;
  }
}

// ---------------- f = conv1x1x1(lrelu(inorm(x))) 64->32 ----------------
__global__ void k_first(const float* __restrict__ x, const float* __restrict__ statsX,
                        const float* __restrict__ w, const float* __restrict__ bias,
                        float* __restrict__ f){
  __shared__ float wsh[CH*CIN];
  __shared__ float msh[CIN], rsh[CIN];
  int b = blockIdx.y;
  for(int i = threadIdx.x; i < CH*CIN; i += blockDim.x) wsh[i] = w[i];
  for(int i = threadIdx.x; i < CIN; i += blockDim.x){
    msh[i] = statsX[(b*CIN+i)*2];
    rsh[i] = rsqrtf(statsX[(b*CIN+i)*2+1] + EPSV);
  }
  __syncthreads();
  int p = blockIdx.x * blockDim.x + threadIdx.x;
  float acc[CH];
  for(int o = 0; o < CH; o++) acc[o] = bias[o];
  for(int ci = 0; ci < CIN; ci++){
    float xn = lrelu((x[((size_t)b*CIN+ci)*NVOX + p] - msh[ci]) * rsh[ci]);
    for(int o = 0; o < CH; o++) acc[o] += wsh[o*CIN+ci] * xn;
  }
  for(int o = 0; o < CH; o++) f[((size_t)b*CH+o)*NVOX + p] = acc[o];
}

// ---------------- fn = lrelu(inorm(f)) elementwise ----------------
__global__ void k_fn(const float* __restrict__ f, const float* __restrict__ st,
                     float* __restrict__ fn){
  int i = blockIdx.x * blockDim.x + threadIdx.x;
  if(i >= BATCH*CH*NVOX) return;
  int ch = i / NVOX;
  fn[i] = lrelu((f[i] - st[ch*2]) * rsqrtf(st[ch*2+1] + EPSV));
}

// -------- stage one 32ch x 16vox B tile for tap t (async LDS if available) --------
__device__ __forceinline__ void stage_tap(const float* __restrict__ srcb,
                                          const float* __restrict__ zbuf,
                                          float* dstbuf, int z, int y, int xbase,
                                          int t, int reflect, int tid, int nthr){
  int tz = t/9 - 1, ty = (t/3)%3 - 1, tx = t%3 - 1;
  for(int q = tid; q < 16*CH; q += nthr){
    int nn = q >> 5, c = q & 31;
    int sz = z + tz, sy = y + ty, sx = xbase + nn + tx;
    const float* g;
    if(reflect){
      sz = sz < 0 ? -sz : (sz > 31 ? 62 - sz : sz);
      sy = sy < 0 ? -sy : (sy > 31 ? 62 - sy : sy);
      sx = sx < 0 ? -sx : (sx > 31 ? 62 - sx : sx);
      g = srcb + (size_t)c*NVOX + ((sz*32+sy)*32 + sx);
    } else {
      bool ok = ((unsigned)sz < 32u) & ((unsigned)sy < 32u) & ((unsigned)sx < 32u);
      g = ok ? (srcb + (size_t)c*NVOX + ((sz*32+sy)*32 + sx)) : zbuf;
    }
#ifdef USE_ASYNC_LDS
    __builtin_amdgcn_global_load_async_to_lds_b32((gas_i32)g, (las_i32)(dstbuf + q), 0, 0);
#else
    dstbuf[q] = *g;
#endif
  }
}

// ------- generic 3x3x3 conv: double-buffered B staging + 27x8 f32 WMMAs -------
// block = 32*mtiles threads (one wave per 16-row M tile), 16 voxels along x.
__global__ void k_conv(const float* __restrict__ src, const float* __restrict__ apack,
                       const float* __restrict__ bias, float* __restrict__ dst,
                       const float* __restrict__ zbuf,
                       int mtiles, int mreal, int reflect){
  __shared__ float bTile[2][16*CH];     // voxel-major: bTile[buf][n*32 + c]
  const int b = blockIdx.z;
  const int zy = blockIdx.y, z = zy >> 5, y = zy & 31;
  const int xbase = blockIdx.x * 16;
  const int tid = threadIdx.x;
  const int lane = tid & 31;
  const int mt = tid >> 5;
  const int n = lane & 15, h = lane >> 4;
  const float* srcb = src + (size_t)b * CH * NVOX;
  v8f acc = {};
  stage_tap(srcb, zbuf, bTile[0], z, y, xbase, 0, reflect, tid, blockDim.x);
  async_wait();
  __syncthreads();
  int cur = 0;
  for(int t = 0; t < 27; t++){
    if(t < 26)   // prefetch next tap's B tile while this tap's WMMAs run
      stage_tap(srcb, zbuf, bTile[cur^1], z, y, xbase, t+1, reflect, tid, blockDim.x);
    const v2f* ap = (const v2f*)(apack + (size_t)(t*mtiles + mt) * 8 * 64);
    const float* brow = bTile[cur] + n*CH;
    for(int s = 0; s < 8; s++){
      v2f a  = ap[s*32 + lane];
      v2f bb = *(const v2f*)(brow + 4*s + 2*h);
      acc = __builtin_amdgcn_wmma_f32_16x16x4_f32(false, a, false, bb,
                                                  (short)0, acc, false, false);
    }
    async_wait();
    __syncthreads();
    cur ^= 1;
  }
  int p = (z*32 + y)*32 + xbase + n;
  for(int r = 0; r < 8; r++){
    int m = mt*16 + r + 8*h;              // C/D layout: lanes16-31 hold M=r+8
    if(m < mreal)
      dst[((size_t)b*mreal + m)*NVOX + p] = acc[r] + (bias ? bias[m] : 0.f);
  }
}

// ------- deformable conv: trilinear-sample B tile into LDS, then WMMA -------
__global__ void k_deform(const float* __restrict__ f, const float* __restrict__ off,
                         const float* __restrict__ apack, float* __restrict__ dst){
  __shared__ float sampB[16*CH];
  __shared__ float fr[3][16];
  __shared__ int   i0[3][16];
  const int b = blockIdx.z;
  const int zy = blockIdx.y, z = zy >> 5, y = zy & 31;
  const int xbase = blockIdx.x * 16;
  const int tid = threadIdx.x;
  const int lane = tid & 31;
  const int mt = tid >> 5;
  const int n = lane & 15, h = lane >> 4;
  const float* fb = f + (size_t)b * CH * NVOX;
  v8f acc = {};
  for(int t = 0; t < 27; t++){
    __syncthreads();
    if(tid < 16){
      int p = (z*32 + y)*32 + xbase + tid;
      float oz = off[((size_t)b*81 + t*3+0)*NVOX + p];
      float oy = off[((size_t)b*81 + t*3+1)*NVOX + p];
      float ox = off[((size_t)b*81 + t*3+2)*NVOX + p];
      float pz = (float)z + (float)(t/9 - 1) + oz;
      float py = (float)y + (float)((t/3)%3 - 1) + oy;
      float px = (float)(xbase + tid) + (float)(t%3 - 1) + ox;
      float fz = floorf(pz), fy = floorf(py), fx = floorf(px);
      i0[0][tid] = (int)fz; i0[1][tid] = (int)fy; i0[2][tid] = (int)fx;
      fr[0][tid] = pz - fz; fr[1][tid] = py - fy; fr[2][tid] = px - fx;
    }
    __syncthreads();
    for(int q = tid; q < 16*CH; q += blockDim.x){
      int nn = q >> 5, c = q & 31;
      int bz = i0[0][nn], by = i0[1][nn], bx = i0[2][nn];
      float gz = fr[0][nn], gy = fr[1][nn], gx = fr[2][nn];
      const float* fc = fb + (size_t)c * NVOX;
      float v = 0.f;
      for(int d = 0; d < 8; d++){
        int dz = (d>>2)&1, dy = (d>>1)&1, dx = d&1;
        int iz = bz+dz, iy = by+dy, ix = bx+dx;
        float w = (dz ? gz : 1.f-gz) * (dy ? gy : 1.f-gy) * (dx ? gx : 1.f-gx);
        if(iz >= 0 && iz < 32 && iy >= 0 && iy < 32 && ix >= 0 && ix < 32)
          v += w * fc[(iz*32+iy)*32 + ix];
      }
      sampB[q] = v;
    }
    __syncthreads();
    const v2f* ap = (const v2f*)(apack + (size_t)(t*2 + mt) * 8 * 64);
    const float* brow = sampB + n*CH;
    for(int s = 0; s < 8; s++){
      v2f a  = ap[s*32 + lane];
      v2f bb = *(const v2f*)(brow + 4*s + 2*h);
      acc = __builtin_amdgcn_wmma_f32_16x16x4_f32(false, a, false, bb,
                                                  (short)0, acc, false, false);
    }
  }
  int p = (z*32 + y)*32 + xbase + n;
  for(int r = 0; r < 8; r++){
    int m = mt*16 + r + 8*h;
    dst[((size_t)b*CH + m)*NVOX + p] = acc[r];
  }
}

// ---------------- comb = belta*dw0 + alpha*dw1 ----------------
__global__ void k_comb(const float* __restrict__ belta, const float* __restrict__ alpha,
                       const float* __restrict__ dw, float* __restrict__ comb){
  int i = blockIdx.x * blockDim.x + threadIdx.x;
  if(i >= BATCH*CH*NVOX) return;
  int b = i / (CH*NVOX);
  comb[i] = belta[i]*dw[b*2] + alpha[i]*dw[b*2+1];
}

// ---------------- out = conv1x1x1(lrelu(inorm(comb))) 32->32 ----------------
__global__ void k_last(const float* __restrict__ comb, const float* __restrict__ st,
                       const float* __restrict__ w, const float* __restrict__ bias,
                       float* __restrict__ out){
  __shared__ float wsh[CH*CH];
  __shared__ float msh[CH], rsh[CH];
  int b = blockIdx.y;
  for(int i = threadIdx.x; i < CH*CH; i += blockDim.x) wsh[i] = w[i];
  for(int i = threadIdx.x; i < CH; i += blockDim.x){
    msh[i] = st[(b*CH+i)*2];
    rsh[i] = rsqrtf(st[(b*CH+i)*2+1] + EPSV);
  }
  __syncthreads();
  int p = blockIdx.x * blockDim.x + threadIdx.x;
  float acc[CH];
  for(int o = 0; o < CH; o++) acc[o] = bias[o];
  for(int ci = 0; ci < CH; ci++){
    float xn = lrelu((comb[((size_t)b*CH+ci)*NVOX + p] - msh[ci]) * rsh[ci]);
    for(int o = 0; o < CH; o++) acc[o] += wsh[o*CH+ci] * xn;
  }
  for(int o = 0; o < CH; o++) out[((size_t)b*CH+o)*NVOX + p] = acc[o];
}

// ---------------- workspace layout (floats) ----------------
static const size_t OFF_STATSX  = 0;         // 256
static const size_t OFF_DW      = 256;       // 4
static const size_t OFF_STATSF  = 512;       // 128
static const size_t OFF_STATSC  = 768;       // 128
static const size_t OFF_ZERO    = 896;       // 64 (zero scratch for async OOB)
static const size_t OFF_PACKOFF = 1024;      // 27*6*8*64 = 82944
static const size_t OFF_PACKNA  = 83968;     // 27*2*8*64 = 27648
static const size_t OFF_PACKDEF = 111616;    // 27648
static const size_t OFF_F       = 139264;    // 2*32*32768
static const size_t OFF_FN      = 2236416;
static const size_t OFF_OFFS    = 4333568;   // 2*81*32768
static const size_t OFF_BELTA   = 9641984;
static const size_t OFF_ALPHA   = 11739136;
static const size_t OFF_COMB    = 13836288;

extern "C" void kernel_launch(void* const* d_in, const int* in_sizes, int n_in,
                              void* d_out, int out_size, void* d_ws, size_t ws_size,
                              hipStream_t stream) {
  (void)in_sizes; (void)n_in; (void)out_size; (void)ws_size;
  const float* x       = (const float*)d_in[0];
  const float* w_first = (const float*)d_in[1];
  const float* b_first = (const float*)d_in[2];
  const float* w_na    = (const float*)d_in[3];
  const float* b_na    = (const float*)d_in[4];
  const float* w_last  = (const float*)d_in[5];
  const float* b_last  = (const float*)d_in[6];
  const float* w_fc1   = (const float*)d_in[7];
  const float* b_fc1   = (const float*)d_in[8];
  const float* w_fc2   = (const float*)d_in[9];
  const float* b_fc2   = (const float*)d_in[10];
  const float* w_off   = (const float*)d_in[11];
  const float* b_off   = (const float*)d_in[12];
  const float* w_def   = (const float*)d_in[13];
  float* out = (float*)d_out;
  float* ws  = (float*)d_ws;

  float* statsX  = ws + OFF_STATSX;
  float* dw      = ws + OFF_DW;
  float* statsF  = ws + OFF_STATSF;
  float* statsC  = ws + OFF_STATSC;
  float* zbuf    = ws + OFF_ZERO;
  float* packOff = ws + OFF_PACKOFF;
  float* packNa  = ws + OFF_PACKNA;
  float* packDef = ws + OFF_PACKDEF;
  float* fbuf    = ws + OFF_F;
  float* fnbuf   = ws + OFF_FN;
  float* offbuf  = ws + OFF_OFFS;
  float* belta   = ws + OFF_BELTA;
  float* alpha   = ws + OFF_ALPHA;
  float* comb    = ws + OFF_COMB;

  // zero scratch, stats of x (also feeds SE gate), gate, weight packing
  k_zero<<<1, 64, 0, stream>>>(zbuf);
  k_stats<<<BATCH*CIN, 256, 0, stream>>>(x, statsX);
  k_gate<<<1, 32, 0, stream>>>(statsX, w_fc1, b_fc1, w_fc2, b_fc2, dw);
  k_pack<<<(27*6*256 + 255)/256, 256, 0, stream>>>(w_off, packOff, 6, 81);
  k_pack<<<(27*2*256 + 255)/256, 256, 0, stream>>>(w_na,  packNa,  2, 32);
  k_pack<<<(27*2*256 + 255)/256, 256, 0, stream>>>(w_def, packDef, 2, 32);

  // f = conv_block(x, w_first) 1x1x1
  k_first<<<dim3(NVOX/128, BATCH), 128, 0, stream>>>(x, statsX, w_first, b_first, fbuf);
  k_stats<<<BATCH*CH, 256, 0, stream>>>(fbuf, statsF);
  k_fn<<<(BATCH*CH*NVOX)/256, 256, 0, stream>>>(fbuf, statsF, fnbuf);

  // off = conv3d(f, w_off, zero pad), WMMA (6 M-tiles for 81 channels)
  k_conv<<<dim3(2, 1024, BATCH), 192, 0, stream>>>(fbuf, packOff, b_off, offbuf, zbuf, 6, 81, 0);
  // belta = deform_conv3d(f, off, w_def), WMMA
  k_deform<<<dim3(2, 1024, BATCH), 64, 0, stream>>>(fbuf, offbuf, packDef, belta);
  // alpha = conv_block(f, w_na, reflect pad), WMMA
  k_conv<<<dim3(2, 1024, BATCH), 64, 0, stream>>>(fnbuf, packNa, b_na, alpha, zbuf, 2, 32, 1);

  // gated combine + final conv_block 1x1x1
  k_comb<<<(BATCH*CH*NVOX)/256, 256, 0, stream>>>(belta, alpha, dw, comb);
  k_stats<<<BATCH*CH, 256, 0, stream>>>(comb, statsC);
  k_last<<<dim3(NVOX/128, BATCH), 128, 0, stream>>>(comb, statsC, w_last, b_last, out);
}